// Center_11184094839250
// MI455X (gfx1250) — compile-verified
//
#include <hip/hip_runtime.h>

// CDNA5 / gfx1250: wave32, WMMA 16x16x4 f32.
typedef __attribute__((ext_vector_type(2))) float v2f;
typedef __attribute__((ext_vector_type(8))) float v8f;

#define N_ROWS   8192
#define N_CLASS  10000
#define SCALE    64.0f

// Deterministic full-wave (32-lane) sum.
// Step 1: one V_WMMA_F32_16X16X4_F32 with A = ones(16x4).
//   B layout (32-bit, 4x16): VGPR0 lanes 0-15 = row K=0, lanes 16-31 = row K=2.
//   With b = {x, 0}, column n of B holds {x_n, 0, x_{n+16}, 0}, so
//   D[m,n] = x_n + x_{n+16} for every m -> every lane ends up holding
//   colsum_{lane&15} in d[0].  (Folds lane n with lane n+16 in one matrix op.)
// Step 2: 4 butterfly shuffles finish 16 -> 1.
// EXEC must be all ones at call sites (WMMA restriction) — both call sites
// below are wave-uniform.
__device__ __forceinline__ float wave_sum_f32(float x) {
    v2f a; a[0] = 1.0f; a[1] = 1.0f;   // A = ones
    v2f b; b[0] = x;    b[1] = 0.0f;   // B carries the lane values
    v8f c = {};
    // 8 args: (neg_a, A, neg_b, B, c_mod, C, reuse_a, reuse_b)
    c = __builtin_amdgcn_wmma_f32_16x16x4_f32(
        false, a, false, b, (short)0, c, false, false);
    float s = c[0];                     // = x_{lane&15} + x_{(lane&15)+16}
    s += __shfl_xor(s, 8);
    s += __shfl_xor(s, 4);
    s += __shfl_xor(s, 2);
    s += __shfl_xor(s, 1);
    return s;                           // every lane holds the wave total
}

__global__ __launch_bounds__(1024) void
center_loss_kernel(const float* __restrict__ feature,
                   const int*   __restrict__ label,
                   float*       __restrict__ out) {
    const int tid = threadIdx.x;

    // Round 1: 8 independent label loads per thread (coalesced per wave).
    int lab[8];
#pragma unroll
    for (int k = 0; k < 8; ++k)
        lab[k] = label[tid + k * 1024];

    // Round 2: 8 independent gathered feature loads per thread.
    float v[8];
#pragma unroll
    for (int k = 0; k < 8; ++k) {
        const long row = (long)(tid + k * 1024);
        v[k] = feature[row * (long)N_CLASS + (long)lab[k]];
    }

    float local = 0.0f;
#pragma unroll
    for (int k = 0; k < 8; ++k)
        local += v[k];

    // Wave-level reduce (WMMA fold + shuffles). Whole block active -> EXEC all 1s.
    const float wsum = wave_sum_f32(local);

    __shared__ float partial[32];
    if ((tid & 31) == 0)
        partial[tid >> 5] = wsum;
    __syncthreads();

    // Wave 0 reduces the 32 per-wave partials (uniform branch: lanes 0-31).
    if (tid < 32) {
        const float total = wave_sum_f32(partial[tid]);
        if (tid == 0) {
            const float loss =
                2.0f - 2.0f * (total / SCALE) / (float)N_ROWS;
            out[0] = loss;
        }
    }
}

extern "C" void kernel_launch(void* const* d_in, const int* in_sizes, int n_in,
                              void* d_out, int out_size, void* d_ws, size_t ws_size,
                              hipStream_t stream) {
    const float* feature = (const float*)d_in[0];
    const int*   label   = (const int*)d_in[1];   // jax int (x64 off -> int32)
    float*       out     = (float*)d_out;

    // Single workgroup (32 waves on one WGP): total HBM traffic ~0.56 MB,
    // so one launch with full load-level parallelism beats any multi-kernel
    // scheme on launch overhead; single block keeps the float-sum order
    // deterministic (no atomics).
    center_loss_kernel<<<dim3(1), dim3(1024), 0, stream>>>(feature, label, out);
}